// EnhancedPrototypeMemory_44100724195854
// MI455X (gfx1250) — compile-verified
//
#include <hip/hip_runtime.h>
#include <hip/hip_bf16.h>

// ---------------------------------------------------------------------------
// EnhancedPrototypeMemory on MI455X (gfx1250, wave32)
//
// Bandwidth-bound problem: stream 134 MB of features once (~5.8 us at
// 23.3 TB/s), normalize rows, segment-sum into 91 class bins. The segment
// sum is done as a one-hot matmul on the matrix pipe via
// V_WMMA_F32_16X16X4_F32 (exact fp32 accumulation: A entries are 0 or
// 1/||x||), keeping the VALU free and avoiding 33.5M LDS atomics.
// ---------------------------------------------------------------------------

typedef __attribute__((ext_vector_type(2))) float v2f;
typedef __attribute__((ext_vector_type(4))) float v4f;
typedef __attribute__((ext_vector_type(8))) float v8f;

#define NUM_CLASSES 91
#define CT 96            // 6 class tiles of 16
#define FEAT 256
#define NROWS 131072
#define TILE 32          // rows per LDS tile
#define BD 512           // 16 waves per block

// output offsets (all outputs emitted as float, concatenated in return order)
#define O_PROTO 0
#define O_VAR   (NUM_CLASSES * FEAT)                 // 23296
#define O_SHAD  (O_VAR + NUM_CLASSES)                // 23387
#define O_INIT  (O_SHAD + NUM_CLASSES * FEAT)        // 46683
#define O_SINIT (O_INIT + NUM_CLASSES)               // 46774
#define O_CNT   (O_SINIT + NUM_CLASSES)              // 46865

__global__ __launch_bounds__(BD)
void epm_accum_kernel(const float* __restrict__ feat,
                      const int* __restrict__ labels,
                      float* __restrict__ wsSums,   // [91*256]
                      float* __restrict__ wsCnt)    // [91]
{
    __shared__ __align__(16) float ldsF[TILE * FEAT]; // 32 KB tile
    __shared__ float ldsInv[TILE];
    __shared__ int   ldsLab[TILE];
    __shared__ int   ldsCnt[CT];

    const int tid      = threadIdx.x;
    const int lane     = tid & 31;
    const int wave     = tid >> 5;        // 0..15 -> owns feature cols [16w,16w+16)
    const int laneHalf = lane >> 4;       // 0/1: K/M split across lane halves
    const int mn       = lane & 15;       // M for A, N for B/C
    const int colBase  = wave * 16;

    if (tid < CT) ldsCnt[tid] = 0;

    v8f acc[6] = {};                      // 6 class tiles x 8 VGPRs (96x16 f32)

    const int numTiles = NROWS / TILE;    // 4096
    v4f stage[4];

    int tile = blockIdx.x;
    if (tile < numTiles) {
        const v4f* gp = (const v4f*)feat + (size_t)tile * (TILE * FEAT / 4);
        #pragma unroll
        for (int s = 0; s < 4; ++s)
            stage[s] = __builtin_nontemporal_load(gp + tid + s * BD); // TH=NT stream
    }

    for (; tile < numTiles; tile += gridDim.x) {
        // commit staged tile to LDS
        #pragma unroll
        for (int s = 0; s < 4; ++s)
            *(v4f*)&ldsF[(tid + s * BD) * 4] = stage[s];
        __syncthreads();

        // ---- norm phase: wave w reduces rows 2w, 2w+1 (wave32 shuffles) ----
        #pragma unroll
        for (int rr = 0; rr < 2; ++rr) {
            int row = wave * 2 + rr;
            float s = 0.0f;
            #pragma unroll
            for (int j = 0; j < 8; ++j) {
                float v = ldsF[row * FEAT + lane + j * 32];
                s += v * v;
            }
            #pragma unroll
            for (int off = 16; off; off >>= 1)
                s += __shfl_xor(s, off, 32);
            if (lane == 0)
                ldsInv[row] = 1.0f / fmaxf(sqrtf(s), 1e-12f);
        }
        if (tid < TILE) {
            int lab = labels[tile * TILE + tid];
            ldsLab[tid] = lab;
            atomicAdd(&ldsCnt[lab], 1);
        }

        // prefetch next tile into registers while WMMA phase runs
        int nt = tile + gridDim.x;
        if (nt < numTiles) {
            const v4f* gp = (const v4f*)feat + (size_t)nt * (TILE * FEAT / 4);
            #pragma unroll
            for (int s = 0; s < 4; ++s)
                stage[s] = __builtin_nontemporal_load(gp + tid + s * BD);
        }
        __syncthreads();

        // ---- WMMA phase: sums += onehot(labels)^T x f_hat, K=4 per op ----
        // A 16x4 layout: lane -> M=mn, VGPR0=K(2*laneHalf), VGPR1=K+1
        // B 4x16 layout: lane -> N=mn, VGPR0=rowK(2*laneHalf), VGPR1=row+1
        #pragma unroll
        for (int kc = 0; kc < TILE; kc += 4) {
            int r0 = kc + 2 * laneHalf;
            int lab0 = ldsLab[r0], lab1 = ldsLab[r0 + 1];
            float in0 = ldsInv[r0], in1 = ldsInv[r0 + 1];
            v2f b;
            b.x = ldsF[r0 * FEAT + colBase + mn];
            b.y = ldsF[(r0 + 1) * FEAT + colBase + mn];
            #pragma unroll
            for (int t = 0; t < 6; ++t) {
                int cls = t * 16 + mn;
                v2f a;
                a.x = (lab0 == cls) ? in0 : 0.0f;  // fold 1/||x|| into one-hot
                a.y = (lab1 == cls) ? in1 : 0.0f;
                acc[t] = __builtin_amdgcn_wmma_f32_16x16x4_f32(
                    false, a, false, b, (short)0, acc[t], false, false);
            }
        }
        __syncthreads();
    }

    // ---- flush: C tile VGPR r holds M=r (lanes 0-15) / M=r+8 (lanes 16-31)
    #pragma unroll
    for (int t = 0; t < 6; ++t) {
        #pragma unroll
        for (int r = 0; r < 8; ++r) {
            int cls = t * 16 + r + 8 * laneHalf;
            if (cls < NUM_CLASSES)
                atomicAdd(&wsSums[cls * FEAT + colBase + mn], acc[t][r]);
        }
    }
    __syncthreads();
    if (tid < NUM_CLASSES)
        atomicAdd(&wsCnt[tid], (float)ldsCnt[tid]);
}

__global__ __launch_bounds__(FEAT)
void epm_finalize_kernel(const float* __restrict__ proto,
                         const unsigned char* __restrict__ pinit,
                         const float* __restrict__ pvar,
                         const float* __restrict__ shadow,
                         const unsigned char* __restrict__ sinit,
                         const int* __restrict__ pcnt,
                         const int* __restrict__ stepp,
                         const float* __restrict__ wsSums,
                         const float* __restrict__ wsCnt,
                         float* __restrict__ out)
{
    const int c = blockIdx.x;
    const int d = threadIdx.x;
    __shared__ float red[FEAT];

    float cnt     = wsCnt[c];
    bool  present = cnt > 0.0f;
    float cls     = wsSums[c * FEAT + d] / fmaxf(cnt, 1.0f);
    float old     = proto[c * FEAT + d];
    bool  pi      = pinit[c] != 0;
    float var     = pvar[c];

    float progress = fminf(1.0f, (float)stepp[0] / 2000.0f); // WARMUP*10
    float mom = pi ? (0.99f + (0.999f - 0.99f) * expf(-var) * progress) : 0.99f;

    float ema  = mom * old + (1.0f - mom) * cls;
    float newp = present ? (pi ? ema : cls) : old;

    float diff = cls - old;
    red[d] = diff * diff;
    __syncthreads();
    #pragma unroll
    for (int s = FEAT / 2; s > 0; s >>= 1) {
        if (d < s) red[d] += red[d + s];
        __syncthreads();
    }
    float upd    = sqrtf(red[0] + 1e-24f);
    float newvar = (present && pi) ? (0.99f * var + 0.01f * upd) : var;

    float sh   = shadow[c * FEAT + d];
    bool  si   = sinit[c] != 0;
    const float shm = 0.9999f;                 // min(0.999+0.0009, 0.9999)
    float shema = shm * sh + (1.0f - shm) * newp;
    float newsh = present ? (si ? shema : newp) : sh;

    out[O_PROTO + c * FEAT + d] = newp;
    out[O_SHAD  + c * FEAT + d] = newsh;
    if (d == 0) {
        out[O_VAR   + c] = newvar;
        out[O_INIT  + c] = (pi || present) ? 1.0f : 0.0f;
        out[O_SINIT + c] = (si || present) ? 1.0f : 0.0f;
        out[O_CNT   + c] = (float)(pcnt[c] + (present ? 1 : 0));
    }
}

extern "C" void kernel_launch(void* const* d_in, const int* in_sizes, int n_in,
                              void* d_out, int out_size, void* d_ws, size_t ws_size,
                              hipStream_t stream) {
    const float*         features = (const float*)d_in[0];
    const int*           labels   = (const int*)d_in[1];
    const float*         protos   = (const float*)d_in[2];
    const unsigned char* pinit    = (const unsigned char*)d_in[3];
    const float*         pvar     = (const float*)d_in[4];
    const float*         shadow   = (const float*)d_in[5];
    const unsigned char* sinit    = (const unsigned char*)d_in[6];
    const int*           pcnt     = (const int*)d_in[7];
    const int*           stepp    = (const int*)d_in[8];

    float* ws     = (float*)d_ws;
    float* wsSums = ws;                          // 91*256 floats
    float* wsCnt  = ws + NUM_CLASSES * FEAT;     // 91 floats

    // zero workspace each launch (graph-capture safe: memset node)
    hipMemsetAsync(d_ws, 0, (size_t)(NUM_CLASSES * FEAT + NUM_CLASSES) * sizeof(float),
                   stream);

    epm_accum_kernel<<<256, BD, 0, stream>>>(features, labels, wsSums, wsCnt);
    epm_finalize_kernel<<<NUM_CLASSES, FEAT, 0, stream>>>(
        protos, pinit, pvar, shadow, sinit, pcnt, stepp, wsSums, wsCnt,
        (float*)d_out);
}